// MLP_55568286875776
// MI455X (gfx1250) — compile-verified
//
#include <hip/hip_runtime.h>
#include <math.h>

typedef __attribute__((ext_vector_type(2))) float v2f;
typedef __attribute__((ext_vector_type(4))) float v4f;
typedef __attribute__((ext_vector_type(8))) float v8f;
typedef unsigned int v4u __attribute__((ext_vector_type(4)));
typedef int v4i __attribute__((ext_vector_type(4)));
typedef int v8i __attribute__((ext_vector_type(8)));

#define LDS_STRIDE 68  // 64 + 4 floats: produced by TDM pad (64 DW interval, 4 DW pad)

// Build a TDM D# for an 8-row x 64-col fp32 tile, row stride 64 floats in
// memory, padded to 68-float rows in LDS. (CDNA5 ISA)
// clang-23 6-arg builtin: (uint32x4 g0, int32x8 g1, int32x4, int32x4, int32x8, Icpol)
template <int CPOL>
__device__ __forceinline__ void tdm_load_8x64(uint64_t gaddr, uint32_t lds_addr) {
    v4u g0;
    g0.x = 1u;                                   // count=1, user mode, no gather
    g0.y = lds_addr;                             // LDS byte address
    g0.z = (uint32_t)gaddr;                      // global_addr[31:0]
    g0.w = (uint32_t)(gaddr >> 32) | (2u << 30); // global_addr[56:32] | type=2

    v8i g1;
    g1[0] = (2 << 16)      // data_size = 4 bytes
          | (1 << 20)      // pad_enable
          | (5 << 22)      // pad_interval: 2^(5+1) = 64 DWORDs between pads
          | (3 << 25);     // pad_amount: 4 DWORDs  -> 68-float LDS rows
    g1[1] = (64 << 16);    // tensor_dim0 = 64 elements
    g1[2] = (8 << 16);     // tensor_dim1 = 8 rows
    g1[3] = (64 << 16);    // tile_dim0 = 64
    g1[4] = 8;             // tile_dim1 = 8, tile_dim2 = 0 (2-D)
    g1[5] = 64;            // tensor_dim0_stride = 64 elements
    g1[6] = 0;
    g1[7] = 0;

    v4i gz4 = {0, 0, 0, 0};              // groups 2/3 unused for 2-D tensors
    v8i gz8 = {0, 0, 0, 0, 0, 0, 0, 0};  // trailing group unused
    __builtin_amdgcn_tensor_load_to_lds(g0, g1, gz4, gz4, gz8, CPOL);
}

__global__ __launch_bounds__(256) void fused_linear_ln_pool(
    const float* __restrict__ x,      // [8192, 64, 64]  (B*N, K, Cin)
    const int*   __restrict__ mask,   // [8192, 64]
    const float* __restrict__ W,      // [64, 64]        (Cout, Cin)
    const float* __restrict__ bvec,   // [64]
    const float* __restrict__ gamma,  // [64]
    const float* __restrict__ beta,   // [64]
    float* __restrict__ out)          // [8192, 64, 128]
{
    __shared__ float Xs[64 * LDS_STRIDE];
    __shared__ float Ws[64 * LDS_STRIDE];
    __shared__ float Ys[64 * LDS_STRIDE];
    __shared__ float pool_s[64];
    __shared__ float bias_s[64], gamma_s[64], beta_s[64];
    __shared__ int   mask_s[64];

    const int bn = blockIdx.x;   // 0..8191
    const int t  = threadIdx.x;  // 0..255

    // ---- Stage X tile and W into LDS via Tensor Data Mover ----
    // One TDM op per wave (TDM ignores EXEC, so never guard it with a lane
    // predicate): wave w moves rows 8w..8w+7. Descriptors are wave-uniform.
    const int wv = __builtin_amdgcn_readfirstlane(t >> 5);   // wave id 0..7
    const float* xsrc = x + (size_t)bn * 4096;
    {
        uint64_t gx = (uint64_t)(uintptr_t)xsrc + (uint64_t)wv * (8 * 64 * 4);
        uint64_t gw = (uint64_t)(uintptr_t)W    + (uint64_t)wv * (8 * 64 * 4);
        uint32_t lx = (uint32_t)(uintptr_t)&Xs[0] + (uint32_t)wv * (8 * LDS_STRIDE * 4);
        uint32_t lw = (uint32_t)(uintptr_t)&Ws[0] + (uint32_t)wv * (8 * LDS_STRIDE * 4);
        tdm_load_8x64<1>(gx, lx);   // TH=NT: x is streamed once
        tdm_load_8x64<0>(gw, lw);   // TH=RT: W stays L2-hot
    }
    if (t < 64) {
        bias_s[t]  = bvec[t];
        gamma_s[t] = gamma[t];
        beta_s[t]  = beta[t];
        mask_s[t]  = mask[(size_t)bn * 64 + t];
    }
    __builtin_amdgcn_s_wait_tensorcnt(0);
    __syncthreads();

    // ---- GEMM via V_WMMA_F32_16X16X4_F32: Y[m,o] = sum_c X[m,c] * W[o,c] ----
    const int lane   = t & 31;
    const int half   = lane >> 4;       // 0/1
    const int l16    = lane & 15;
    const int mt     = wv & 3;          // 16-row band of M (neighbor dim)
    const int ntbase = (wv >> 2) * 2;   // this wave covers N bands ntbase, ntbase+1

    v8f acc0 = {};
    v8f acc1 = {};
    const float* Arow  = &Xs[(mt * 16 + l16) * LDS_STRIDE];
    const float* Brow0 = &Ws[((ntbase + 0) * 16 + l16) * LDS_STRIDE];
    const float* Brow1 = &Ws[((ntbase + 1) * 16 + l16) * LDS_STRIDE];
#pragma unroll
    for (int kk = 0; kk < 16; ++kk) {
        int c = kk * 4 + half * 2;      // ISA layout: VGPR0=K0|K2, VGPR1=K1|K3
        v2f a  = { Arow[c],  Arow[c + 1]  };
        v2f b0 = { Brow0[c], Brow0[c + 1] };
        v2f b1 = { Brow1[c], Brow1[c + 1] };
        acc0 = __builtin_amdgcn_wmma_f32_16x16x4_f32(
            false, a, false, b0, (short)0, acc0, false, false);
        acc1 = __builtin_amdgcn_wmma_f32_16x16x4_f32(
            false, a, false, b1, (short)0, acc1, false, false);
    }

    // D layout: VGPR r, lanes 0-15 -> row r, lanes 16-31 -> row r+8; col = l16
#pragma unroll
    for (int r = 0; r < 8; ++r) {
        int row  = mt * 16 + r + half * 8;
        int col0 = ntbase * 16 + l16;
        Ys[row * LDS_STRIDE + col0]      = acc0[r] + bias_s[col0];
        Ys[row * LDS_STRIDE + col0 + 16] = acc1[r] + bias_s[col0 + 16];
    }
    __syncthreads();

    // ---- Per-row LayerNorm + LeakyReLU (threads 0..63, one row each) ----
    if (t < 64) {
        float* yr = &Ys[t * LDS_STRIDE];
        float s = 0.f;
#pragma unroll
        for (int c = 0; c < 64; ++c) s += yr[c];
        float mu = s * (1.0f / 64.0f);
        float var = 0.f;
#pragma unroll
        for (int c = 0; c < 64; ++c) { float d = yr[c] - mu; var += d * d; }
        float rstd = rsqrtf(var * (1.0f / 64.0f) + 1e-5f);
#pragma unroll
        for (int c = 0; c < 64; ++c) {
            float yv = (yr[c] - mu) * rstd * gamma_s[c] + beta_s[c];
            yr[c] = yv >= 0.f ? yv : 0.01f * yv;
        }
    }
    __syncthreads();

    // ---- Masked max-pool over the 64 neighbor rows, per column ----
    if (t < 64) {
        float m = -INFINITY;
        for (int r = 0; r < 64; ++r) {
            if (mask_s[r]) m = fmaxf(m, Ys[r * LDS_STRIDE + t]);
        }
        pool_s[t] = m;
    }
    __syncthreads();

    // ---- Coalesced non-temporal float4 stores of [64 x 128] concat block ----
    // 256 MB pure output stream, never re-read: NT keeps it out of the 192 MB L2.
    float* orow = out + (size_t)bn * (64 * 128);
#pragma unroll
    for (int i = 0; i < 8; ++i) {
        int f = t * 4 + i * 1024;       // 0..8191, groups of 4 never straddle col 64
        int r = f >> 7;
        int c = f & 127;
        v4f v;
        if (c < 64) {
            bool mk = mask_s[r] != 0;
            v.x = mk ? Ys[r * LDS_STRIDE + c + 0] : 0.f;
            v.y = mk ? Ys[r * LDS_STRIDE + c + 1] : 0.f;
            v.z = mk ? Ys[r * LDS_STRIDE + c + 2] : 0.f;
            v.w = mk ? Ys[r * LDS_STRIDE + c + 3] : 0.f;
        } else {
            v.x = pool_s[c - 64];
            v.y = pool_s[c - 63];
            v.z = pool_s[c - 62];
            v.w = pool_s[c - 61];
        }
        __builtin_nontemporal_store(v, (v4f*)(orow + f));
    }
}

extern "C" void kernel_launch(void* const* d_in, const int* in_sizes, int n_in,
                              void* d_out, int out_size, void* d_ws, size_t ws_size,
                              hipStream_t stream) {
    const float* x     = (const float*)d_in[0];
    const int*   mask  = (const int*)d_in[1];
    const float* W     = (const float*)d_in[2];
    const float* b     = (const float*)d_in[3];
    const float* gamma = (const float*)d_in[4];
    const float* beta  = (const float*)d_in[5];
    float* out = (float*)d_out;

    const int BN = 8 * 1024;  // one workgroup per (b, n)
    fused_linear_ln_pool<<<dim3(BN), dim3(256), 0, stream>>>(
        x, mask, W, b, gamma, beta, out);
}